// Arcface_48782238548531
// MI455X (gfx1250) — compile-verified
//
#include <hip/hip_runtime.h>
#include <math.h>

typedef float v2f __attribute__((ext_vector_type(2)));
typedef float v4f __attribute__((ext_vector_type(4)));
typedef float v8f __attribute__((ext_vector_type(8)));

#define EMBED     128
#define CLASSNUM  100000
#define NB        2048
#define SCALE_S   64.0f
#define COS_M_F   0.8775825618903728f   // cos(0.5)
#define SIN_M_F   0.4794255386042030f   // sin(0.5)

#define M_BLK      64
#define N_BLK      160      // 2 waves x 5 tiles x 16
#define LDS_STRIDE 132      // 128 + 4 pad floats -> conflict-free fragment reads

// ---------------------------------------------------------------------------
// Pass 1: reciprocal column norms of W[EMBED][CLASSNUM]
// ---------------------------------------------------------------------------
__global__ void arcface_colnorm(const float* __restrict__ W,
                                float* __restrict__ rnorm) {
    int n = blockIdx.x * blockDim.x + threadIdx.x;
    if (n >= CLASSNUM) return;
    float s = 0.0f;
#pragma unroll 8
    for (int k = 0; k < EMBED; ++k) {
        float w = W[(size_t)k * CLASSNUM + n];
        s += w * w;
    }
    rnorm[n] = 1.0f / sqrtf(s);
}

// ---------------------------------------------------------------------------
// Pass 2: WMMA GEMM (E @ W) with fused normalize / clip / arcface epilogue
// ---------------------------------------------------------------------------
__global__ void __launch_bounds__(256, 2)
arcface_gemm(const float* __restrict__ E,      // [NB][EMBED] row-major
             const int*   __restrict__ label,  // [NB]
             const float* __restrict__ W,      // [EMBED][CLASSNUM] row-major
             const float* __restrict__ rnorm,  // [CLASSNUM]
             float* __restrict__ out)          // [NB][CLASSNUM]
{
    __shared__ float Alds[M_BLK * LDS_STRIDE];

    const int tid   = threadIdx.x;
    const int lane  = tid & 31;
    const int wid   = tid >> 5;                 // 8 waves
    const int m_blk = blockIdx.y * M_BLK;
    const int n_blk = blockIdx.x * N_BLK;

    // --- cooperative A-block load: 64 rows x 128 cols (float4 per thread) ---
    for (int i = tid; i < M_BLK * (EMBED / 4); i += 256) {
        int row = i >> 5;                       // EMBED/4 == 32 float4 per row
        int c4  = i & 31;
        v4f v = *(const v4f*)(E + (size_t)(m_blk + row) * EMBED + c4 * 4);
        *(v4f*)(&Alds[row * LDS_STRIDE + c4 * 4]) = v;
    }
    __syncthreads();

    // wave -> 16(M) x 80(N) strip
    const int wrow  = wid & 3;                  // 4 wave-rows
    const int wcol  = wid >> 2;                 // 2 wave-cols
    const int m_off = wrow * 16;
    const int n0    = n_blk + wcol * 80;

    const int lhalf = lane >> 4;                // 0: lanes 0-15, 1: lanes 16-31
    const int l15   = lane & 15;
    const int kb    = lhalf * 2;                // A/B fragment K sub-offset

    v8f acc[5] = {};                            // 5 x (16x16 f32) accumulators

    const float* Arow = &Alds[(m_off + l15) * LDS_STRIDE];

    for (int k = 0; k < EMBED; k += 4) {
        // A fragment: lane l15 = row, VGPR pair = K {k+kb, k+kb+1}
        v2f a = *(const v2f*)(Arow + k + kb);
        const float* Bbase = W + (size_t)(k + kb) * CLASSNUM + n0 + l15;
#pragma unroll
        for (int t = 0; t < 5; ++t) {
            v2f b;
            b.x = Bbase[t * 16];                // B[k+kb  ][col]
            b.y = Bbase[t * 16 + CLASSNUM];     // B[k+kb+1][col]
            acc[t] = __builtin_amdgcn_wmma_f32_16x16x4_f32(
                false, a, false, b, (short)0, acc[t], false, false);
        }
    }

    // --- epilogue: normalize, clip, arcface margin at label, scale, store ---
    int labr[8];
#pragma unroll
    for (int r = 0; r < 8; ++r)
        labr[r] = label[m_blk + m_off + lhalf * 8 + r];

#pragma unroll
    for (int t = 0; t < 5; ++t) {
        int   col = n0 + t * 16 + l15;
        float rn  = rnorm[col];
        float* outb = out + (size_t)(m_blk + m_off + lhalf * 8) * CLASSNUM + col;
#pragma unroll
        for (int r = 0; r < 8; ++r) {
            float c = acc[t][r] * rn;
            c = fminf(fmaxf(c, -1.0f), 1.0f);
            float v = c * SCALE_S;
            if (labr[r] == col) {
                float s = sqrtf(fmaxf(1.0f - c * c, 0.0f));
                v = (c * COS_M_F - s * SIN_M_F) * SCALE_S;
            }
            outb[(size_t)r * CLASSNUM] = v;
        }
    }
}

// ---------------------------------------------------------------------------
extern "C" void kernel_launch(void* const* d_in, const int* in_sizes, int n_in,
                              void* d_out, int out_size, void* d_ws, size_t ws_size,
                              hipStream_t stream) {
    const float* E     = (const float*)d_in[0];   // embbedings [NB][EMBED]
    const int*   label = (const int*)  d_in[1];   // label      [NB]
    const float* W     = (const float*)d_in[2];   // kernel     [EMBED][CLASSNUM]
    float*       out   = (float*)d_out;           // [NB][CLASSNUM]
    float*       rnorm = (float*)d_ws;            // CLASSNUM floats scratch

    arcface_colnorm<<<(CLASSNUM + 255) / 256, 256, 0, stream>>>(W, rnorm);

    dim3 grid(CLASSNUM / N_BLK, NB / M_BLK);      // 625 x 32 = 20000 blocks
    arcface_gemm<<<grid, dim3(256), 0, stream>>>(E, label, W, rnorm, out);
}